// SparseMixerMoeRoutingMethod_25572235280541
// MI455X (gfx1250) — compile-verified
//
#include <hip/hip_runtime.h>
#include <hip/hip_bf16.h>
#include <cstdint>

#define NEXPERTS 64
#define TPB      256           // threads per block == tokens per tile
#define EPS2     0.2f          // 2 * EPS

// CDNA5 async global->LDS DMA path (ASYNCcnt-tracked), with fallback so the
// compile loop reports which path lowered.
#if defined(__HIP_DEVICE_COMPILE__) &&                                         \
    __has_builtin(__builtin_amdgcn_global_load_async_to_lds_b128) &&           \
    __has_builtin(__builtin_amdgcn_s_wait_asynccnt)
#define USE_ASYNC_LDS 1
#else
#define USE_ASYNC_LDS 0
#endif

// Builtin signature is (global v4i*, local v4i*, imm int offset, imm int cpol)
typedef int v4i __attribute__((__vector_size__(16)));
typedef __attribute__((address_space(1))) v4i gbl_v4i;
typedef __attribute__((address_space(3))) v4i lds_v4i;

__device__ __forceinline__ void stage_chunk(const float* gsrc, float4* ldst) {
#if USE_ASYNC_LDS
  __builtin_amdgcn_global_load_async_to_lds_b128(
      (gbl_v4i*)(uintptr_t)gsrc,
      (lds_v4i*)(uint32_t)(uintptr_t)ldst,   // generic->AS3: low 32 bits are the LDS offset
      0, 0);
#else
  *ldst = *(const float4*)gsrc;
#endif
}

__global__ __launch_bounds__(TPB) void
sparse_mixer_top2_kernel(const float* __restrict__ logits,
                         int2* __restrict__ out_idx,
                         float2* __restrict__ out_val,
                         int ntok) {
  // XOR-swizzled tile: row r, 16B chunk c lives at tile[r*16 + (c ^ (r&15) ^ ((r&16)>>1))].
  // Swizzle keeps per-token ds_load_b128 reads at the 2-cycle bank floor.
  __shared__ float4 tile[TPB * (NEXPERTS / 4)];   // 64 KB

  const unsigned t    = threadIdx.x;
  const int      tok0 = (int)blockIdx.x * TPB;

  // ---------------- stage: global -> LDS (4 KB coalesced bursts) ----------------
  const unsigned cchunk = t & 15u;   // 16B chunk-column this thread fetches
  const unsigned rsub   = t >> 4;    // row sub-index 0..15
  const float*   gtile  = logits + (size_t)tok0 * NEXPERTS;   // uniform 64-bit base

  if (tok0 + TPB <= ntok) {
    // Full tile (the common case: 524288 % 256 == 0) — no per-lane exec games.
#pragma unroll
    for (unsigned k = 0; k < 16u; ++k) {
      const unsigned row = k * 16u + rsub;                     // 0..255
      const unsigned sw  = cchunk ^ (row & 15u) ^ ((row & 16u) >> 1);
      stage_chunk(gtile + (row * (unsigned)NEXPERTS + cchunk * 4u),
                  &tile[row * 16u + sw]);
    }
  } else {
    // Ragged tail tile: per-lane bound check.
#pragma unroll
    for (unsigned k = 0; k < 16u; ++k) {
      const unsigned row = k * 16u + rsub;
      const unsigned sw  = cchunk ^ (row & 15u) ^ ((row & 16u) >> 1);
      if (tok0 + (int)row < ntok) {
        stage_chunk(gtile + (row * (unsigned)NEXPERTS + cchunk * 4u),
                    &tile[row * 16u + sw]);
      }
    }
  }
#if USE_ASYNC_LDS
  __builtin_amdgcn_s_wait_asynccnt(0);   // each wave drains its own async copies
#endif
  __syncthreads();                       // then all waves see the full tile

  // ---------------- compute: one thread per token ----------------
  const int token = tok0 + (int)t;
  if (token < ntok) {
    float v[NEXPERTS];
#pragma unroll
    for (unsigned c = 0; c < 16u; ++c) {
      const unsigned sw = c ^ (t & 15u) ^ ((t & 16u) >> 1);
      const float4 q = tile[t * 16u + sw];
      v[c * 4 + 0] = q.x; v[c * 4 + 1] = q.y;
      v[c * 4 + 2] = q.z; v[c * 4 + 3] = q.w;
    }

    // Top-2 with first-occurrence tie-break (strict > matches jnp.argmax).
    float m1 = v[0];               int i1 = 0;
    float m2 = -__builtin_inff();  int i2 = 0;
#pragma unroll
    for (int j = 1; j < NEXPERTS; ++j) {
      const float x  = v[j];
      const bool  g1 = x > m1;
      const bool  g2 = x > m2;
      const float nm2 = g1 ? m1 : (g2 ? x : m2);
      const int   ni2 = g1 ? i1 : (g2 ? j : i2);
      m1 = g1 ? x : m1;
      i1 = g1 ? j : i1;
      m2 = nm2;
      i2 = ni2;
    }

    // Masked-softmax denominators. mask: (m - x)/max(|x|, m) > 0.2, division-free
    // (sign/NaN cases agree with the JAX formula since max(|x|, m) >= 0).
    // Pass-2 reuses pass-1 exponentials: sum exp(x - m2) = exp(m1 - m2) * sum exp(x - m1).
    float s1 = 0.0f, s2 = 0.0f;
#pragma unroll
    for (int j = 0; j < NEXPERTS; ++j) {
      const float x = v[j];
      const float e = __expf(x - m1);
      const float a = fabsf(x);
      const bool drop1 = (m1 - x) > EPS2 * fmaxf(a, m1);
      const bool drop2 = ((m2 - x) > EPS2 * fmaxf(a, m2)) || (j == i1);
      s1 += drop1 ? 0.0f : e;
      s2 += drop2 ? 0.0f : e;
    }

    out_idx[token] = make_int2(i1, i2);                                   // raw int32 bits
    out_val[token] = make_float2(1.0f / s1, __expf(m2 - m1) / s2);
  }
}

extern "C" void kernel_launch(void* const* d_in, const int* in_sizes, int n_in,
                              void* d_out, int out_size, void* d_ws, size_t ws_size,
                              hipStream_t stream) {
  (void)n_in; (void)out_size; (void)d_ws; (void)ws_size;
  const float* logits = (const float*)d_in[0];
  const int    ntok   = in_sizes[0] / NEXPERTS;

  // d_out layout: [ntok,2] int32 indices (raw bits), then [ntok,2] fp32 values.
  int2*   out_idx = (int2*)d_out;
  float2* out_val = (float2*)((float*)d_out + (size_t)ntok * 2);

  const int blocks = (ntok + TPB - 1) / TPB;
  sparse_mixer_top2_kernel<<<blocks, TPB, 0, stream>>>(logits, out_idx, out_val, ntok);
}